// PCDPretreatment_88235808129103
// MI455X (gfx1250) — compile-verified
//
#include <hip/hip_runtime.h>
#include <hip/hip_bf16.h>
#include <math.h>

typedef unsigned long long u64;
typedef unsigned int u32;

#define NB    64          // persistent blocks (must be co-resident)
#define NT    256         // threads per block (8 wave32 waves)
#define NW    (NT / 32)   // waves per block
#define PPT   16          // points per thread: 64*256*16 = 262144
#define NPTS  262144
#define NSAMP 4096

// workspace layout (first 4 KB zeroed each launch via hipMemsetAsync)
#define WS_COUNTER 0                       // u64 arrival counter
#define WS_PSUM    128                     // float4 psum[NB]
#define WS_PMAX    2176                    // float  pmax[NB]
#define WS_KEYS    2560                    // u64    keys[2][NB]
#define WS_INIT_BYTES 4096

__device__ __forceinline__ u64 umax64(u64 a, u64 b) { return (b > a) ? b : a; }

__device__ __forceinline__ void grid_sync(u64* counter, u64 target) {
  __syncthreads();
  if (threadIdx.x == 0) {
    __hip_atomic_fetch_add(counter, 1ull, __ATOMIC_ACQ_REL, __HIP_MEMORY_SCOPE_AGENT);
    while (__hip_atomic_load(counter, __ATOMIC_ACQUIRE, __HIP_MEMORY_SCOPE_AGENT) < target) {
      __builtin_amdgcn_s_sleep(1);
    }
  }
  __syncthreads();
  __threadfence();   // device-scope acquire for all threads in the block
}

__global__ __launch_bounds__(NT)
void fps_persistent_kernel(const float* __restrict__ in,
                           float* __restrict__ out,
                           char* __restrict__ ws) {
  const u32 tid  = threadIdx.x;
  const u32 bid  = blockIdx.x;
  const u32 lane = tid & 31u;
  const u32 wid  = tid >> 5;

  u64*    counter = (u64*)(ws + WS_COUNTER);
  float4* psum    = (float4*)(ws + WS_PSUM);
  float*  pmax    = (float*)(ws + WS_PMAX);
  u64*    gkeys   = (u64*)(ws + WS_KEYS);

  __shared__ alignas(16) char stage[NT * PPT * 12];  // 48 KB tile staging
  __shared__ float4 lf[NT];
  __shared__ u64    lu[NW];
  __shared__ u64    lwin;

  // ---- stage this block's 4096-point tile via CDNA5 async global->LDS ----
  {
    u64 gsrc  = (u64)((const char*)in + (size_t)bid * (NT * PPT * 12)
                                      + (size_t)tid * (PPT * 12));
    u32 lbase = (u32)(size_t)(&stage[tid * (PPT * 12)]);
#pragma unroll
    for (int k = 0; k < (PPT * 12) / 16; ++k) {     // 12 x b128 per thread
      u32 laddr = lbase + 16u * (u32)k;
      u64 gaddr = gsrc + 16u * (u32)k;
      asm volatile("global_load_async_to_lds_b128 %0, %1, off"
                   :: "v"(laddr), "v"(gaddr) : "memory");
    }
    asm volatile("s_wait_asynccnt 0" ::: "memory");
  }
  __syncthreads();

  // ---- pull this thread's 16 points into registers ----
  float px[PPT], py[PPT], pz[PPT], dist[PPT];
  {
    const float* sp = (const float*)(stage + tid * (PPT * 12));
#pragma unroll
    for (int i = 0; i < PPT; ++i) {
      px[i] = sp[3 * i + 0];
      py[i] = sp[3 * i + 1];
      pz[i] = sp[3 * i + 2];
    }
  }

  // ---- phase 0: deterministic tree-sum -> mean (runs once; LDS tree) ----
  float sx = 0.f, sy = 0.f, sz = 0.f;
#pragma unroll
  for (int i = 0; i < PPT; ++i) { sx += px[i]; sy += py[i]; sz += pz[i]; }
  lf[tid] = make_float4(sx, sy, sz, 0.f);
  __syncthreads();
  for (int s = NT / 2; s > 0; s >>= 1) {
    if (tid < s) {
      float4 a = lf[tid], b = lf[tid + s];
      lf[tid] = make_float4(a.x + b.x, a.y + b.y, a.z + b.z, 0.f);
    }
    __syncthreads();
  }
  if (tid == 0) psum[bid] = lf[0];
  grid_sync(counter, (u64)NB * 1);

  {
    float4 v = (tid < NB) ? psum[tid] : make_float4(0.f, 0.f, 0.f, 0.f);
    lf[tid] = v;
    __syncthreads();
    for (int s = NT / 2; s > 0; s >>= 1) {
      if (tid < s) {
        float4 a = lf[tid], b = lf[tid + s];
        lf[tid] = make_float4(a.x + b.x, a.y + b.y, a.z + b.z, 0.f);
      }
      __syncthreads();
    }
  }
  const float meanx = lf[0].x * (1.0f / (float)NPTS);
  const float meany = lf[0].y * (1.0f / (float)NPTS);
  const float meanz = lf[0].z * (1.0f / (float)NPTS);

  // ---- phase 1: max squared norm -> max_dis (runs once; LDS tree) ----
  float mm = 0.f;
#pragma unroll
  for (int i = 0; i < PPT; ++i) {
    float dx = px[i] - meanx, dy = py[i] - meany, dz = pz[i] - meanz;
    mm = fmaxf(mm, dx * dx + dy * dy + dz * dz);
  }
  __syncthreads();                       // lf[0] was just read by everyone
  lf[tid] = make_float4(mm, 0.f, 0.f, 0.f);
  __syncthreads();
  for (int s = NT / 2; s > 0; s >>= 1) {
    if (tid < s) lf[tid].x = fmaxf(lf[tid].x, lf[tid + s].x);
    __syncthreads();
  }
  if (tid == 0) pmax[bid] = lf[0].x;
  grid_sync(counter, (u64)NB * 2);

  {
    float v = (tid < NB) ? pmax[tid] : 0.f;
    lf[tid] = make_float4(v, 0.f, 0.f, 0.f);
    __syncthreads();
    for (int s = NT / 2; s > 0; s >>= 1) {
      if (tid < s) lf[tid].x = fmaxf(lf[tid].x, lf[tid + s].x);
      __syncthreads();
    }
  }
  const float maxd = sqrtf(lf[0].x);

  // ---- normalize register-resident points; init dist ----
#pragma unroll
  for (int i = 0; i < PPT; ++i) {
    px[i] = (px[i] - meanx) / maxd;
    py[i] = (py[i] - meany) / maxd;
    pz[i] = (pz[i] - meanz) / maxd;
    dist[i] = 1e10f;
  }
  __syncthreads();   // retire lf before the hot loop

  // ---- phase 2: 4096 sequential FPS iterations (latency-optimized) ----
  u32 f = 0;   // idx[0] = 0, matching the reference's deterministic start
  for (int t = 0; t < NSAMP; ++t) {
    // centroid = normalized coords of point f (broadcast load; L2 hit)
    const float cx = (in[3 * (size_t)f + 0] - meanx) / maxd;
    const float cy = (in[3 * (size_t)f + 1] - meany) / maxd;
    const float cz = (in[3 * (size_t)f + 2] - meanz) / maxd;

    // output column t is exactly this centroid (pcd.T layout: 3 x 4096)
    if (bid == 0 && tid == 0) {
      out[t]             = cx;
      out[NSAMP + t]     = cy;
      out[2 * NSAMP + t] = cz;
    }

    // update running min-distances; pack argmax key:
    // (dist_bits << 32) | (~index)  -> max key == max dist, tie -> first index
    u64 best = 0ull;
#pragma unroll
    for (int i = 0; i < PPT; ++i) {
      float dx = px[i] - cx, dy = py[i] - cy, dz = pz[i] - cz;
      float d  = fmaf(dx, dx, fmaf(dy, dy, dz * dz));
      dist[i]  = fminf(dist[i], d);
      u32 gi   = bid * (NT * PPT) + tid * PPT + (u32)i;
      u64 key  = ((u64)__float_as_uint(dist[i]) << 32) | (u64)(0xFFFFFFFFu - gi);
      best     = umax64(best, key);
    }

    // wave32 butterfly argmax (no barriers), then 8-wave combine (1 barrier)
#pragma unroll
    for (int off = 16; off > 0; off >>= 1)
      best = umax64(best, __shfl_xor(best, off, 32));
    if (lane == 0) lu[wid] = best;
    __syncthreads();
    u64 blockbest = lu[0];
#pragma unroll
    for (int w = 1; w < NW; ++w) blockbest = umax64(blockbest, lu[w]);
    if (tid == 0) gkeys[(t & 1) * NB + bid] = blockbest;  // double-buffered

    grid_sync(counter, (u64)NB * (u64)(t + 3));

    // cross-block argmax: wave 0 covers all 64 published keys (2 per lane)
    if (wid == 0) {
      const u64* kb = gkeys + (t & 1) * NB;
      u64 m = umax64(kb[lane], kb[lane + 32]);
#pragma unroll
      for (int off = 16; off > 0; off >>= 1)
        m = umax64(m, __shfl_xor(m, off, 32));
      if (lane == 0) lwin = m;
    }
    __syncthreads();
    f = 0xFFFFFFFFu - (u32)(lwin & 0xFFFFFFFFull);
    __syncthreads();   // protect lwin/lu before next iteration's writes
  }
}

extern "C" void kernel_launch(void* const* d_in, const int* in_sizes, int n_in,
                              void* d_out, int out_size, void* d_ws, size_t ws_size,
                              hipStream_t stream) {
  (void)in_sizes; (void)n_in; (void)out_size; (void)ws_size;
  const float* in = (const float*)d_in[0];
  float* out = (float*)d_out;

  // reset arrival counter / partial buffers every call (graph-capture legal)
  hipMemsetAsync(d_ws, 0, WS_INIT_BYTES, stream);

  hipLaunchKernelGGL(fps_persistent_kernel, dim3(NB), dim3(NT), 0, stream,
                     in, out, (char*)d_ws);
}